// SSM_60816736911668
// MI455X (gfx1250) — compile-verified
//
#include <hip/hip_runtime.h>
#include <hip/hip_bf16.h>

typedef __attribute__((ext_vector_type(2))) float v2f;
typedef __attribute__((ext_vector_type(8))) float v8f;
typedef __attribute__((ext_vector_type(2))) int v2i;

#define BATCH 8
#define SEQL  2048
#define DDIM  64
#define NDIM  64

#if defined(__gfx1250__) && __has_builtin(__builtin_amdgcn_global_load_async_to_lds_b64)
#define SSM_ASYNC_LDS 1
#else
#define SSM_ASYNC_LDS 0
#endif

// Stage 8 contiguous bytes of x into LDS. CDNA5 path: async DMA to LDS
// (tracked by ASYNCcnt, no VGPR round-trip); fallback: regular load/store.
// Probe-derived builtin signature: (AS1 v2i* gsrc, AS3 v2i* ldst, Ii, Ii).
__device__ __forceinline__ void stage_b64(const float* __restrict__ g,
                                          float* __restrict__ l) {
#if SSM_ASYNC_LDS
  __builtin_amdgcn_global_load_async_to_lds_b64(
      (__attribute__((address_space(1))) v2i*)g,
      (__attribute__((address_space(3))) v2i*)l, 0, 0);
#else
  v2f v = *(const v2f*)g;
  l[0] = v.x;
  l[1] = v.y;
#endif
}

__device__ __forceinline__ void stage_wait() {
#if SSM_ASYNC_LDS
#if __has_builtin(__builtin_amdgcn_s_wait_asynccnt)
  __builtin_amdgcn_s_wait_asynccnt(0);
#else
  asm volatile("s_wait_asynccnt 0" ::: "memory");
#endif
#endif
}

// 64x64x64 f32 matmul by one 256-thread (8-wave) workgroup using
// V_WMMA_F32_16X16X4_F32.  Dst = alpha * (A @ B) (+ I if addI).
// Operand layouts per CDNA5 ISA 7.12.2 (wave32):
//   A 16x4 f32 : lanes 0-15 -> M=lane, vgpr0=K0,vgpr1=K1; lanes16-31 -> K2,K3
//   B 4x16 f32 : lanes hold N=lane%16; vgpr0=K0(/K2), vgpr1=K1(/K3)
//   C/D 16x16  : vgpr i -> M=i (lanes 0-15) / M=i+8 (lanes 16-31), N=lane%16
__device__ __forceinline__ void mm64(const float* __restrict__ A, int sA,
                                     const float* __restrict__ B, int sB,
                                     float* __restrict__ Dst, int sD,
                                     float alpha, bool addI) {
  const int tid  = threadIdx.x;
  const int wave = tid >> 5;
  const int lane = tid & 31;
  const int hl   = lane >> 4;   // which 16-lane half
  const int r    = lane & 15;
  for (int t = wave; t < 16; t += 8) {
    const int tm = t >> 2, tn = t & 3;
    v8f acc = {};
#pragma unroll
    for (int kk = 0; kk < 16; ++kk) {
      const int k0 = kk * 4 + 2 * hl;
      v2f af, bf;
      af.x = A[(tm * 16 + r) * sA + k0 + 0];
      af.y = A[(tm * 16 + r) * sA + k0 + 1];
      bf.x = B[(k0 + 0) * sB + tn * 16 + r];
      bf.y = B[(k0 + 1) * sB + tn * 16 + r];
      acc = __builtin_amdgcn_wmma_f32_16x16x4_f32(false, af, false, bf,
                                                  (short)0, acc, false, false);
    }
#pragma unroll
    for (int i = 0; i < 8; ++i) {
      const int row = tm * 16 + hl * 8 + i;
      const int col = tn * 16 + r;
      float v = acc[i] * alpha;
      if (addI && row == col) v += 1.0f;
      Dst[row * sD + col] = v;
    }
  }
}

// Kernel 1 (single block): compute dA = expm(deltaA) via scaling+squaring
// Taylor (all matmuls on WMMA), then A_inv_term = (deltaA+1e-5 I)^-1 (dA - I)
// via Gauss-Jordan with partial pivoting in LDS, then dB = A_inv_term @
// (dtc[:,None]*B).  Writes dA to ws[0..4095], dB to ws[4096..8191].
__global__ __launch_bounds__(256) void ssm_setup(
    const float* __restrict__ Bm, const float* __restrict__ dt,
    const float* __restrict__ A_log, float* __restrict__ ws) {
  __shared__ float smem[12288];   // 48 KB = three 64x64 f32 tiles
  __shared__ float rowsum[64];
  __shared__ float fac[64];
  __shared__ int piv_sh;
  __shared__ int s_sh;
  float* bufA = smem;             // Ms = scaled deltaA
  float* bufB = smem + 4096;
  float* bufC = smem + 8192;
  float* aug  = smem;             // 64x128 augmented system (aliases A+B)
  const int tid = threadIdx.x;

  // ---- inf-norm of deltaA -> squaring count s ------------------------
  if (tid < 64) {
    const float dtc = fminf(fmaxf(dt[tid], -0.1f), 0.1f);
    float s = 0.f;
    for (int j = 0; j < 64; ++j) s += fabsf(dtc * expf(A_log[tid * 64 + j]));
    rowsum[tid] = s;
  }
  __syncthreads();
  if (tid == 0) {
    float nrm = 0.f;
    for (int i = 0; i < 64; ++i) nrm = fmaxf(nrm, rowsum[i]);
    int s = 0;
    while (nrm > 0.25f && s < 32) { nrm *= 0.5f; ++s; }
    s_sh = s;
  }
  __syncthreads();
  const int sexp = s_sh;
  const float scale = exp2f((float)(-sexp));

  // ---- Ms = deltaA * 2^-s (bufA); T = I (bufC) -----------------------
  for (int e = tid; e < 4096; e += 256) {
    const int i = e >> 6, j = e & 63;
    const float dtc = fminf(fmaxf(dt[i], -0.1f), 0.1f);
    bufA[e] = dtc * (-expf(A_log[e])) * scale;
    bufC[e] = (i == j) ? 1.0f : 0.0f;
  }
  __syncthreads();

  // ---- Taylor (Horner, K=10; ||Ms|| <= 0.25 -> err ~1e-14) -----------
  float* Ts = bufC;
  float* Td = bufB;
  for (int k = 10; k >= 1; --k) {
    mm64(bufA, 64, Ts, 64, Td, 64, 1.0f / (float)k, true);
    __syncthreads();
    float* t = Ts; Ts = Td; Td = t;
  }  // K even -> result in bufC

  // ---- squaring ------------------------------------------------------
  for (int j = 0; j < sexp; ++j) {
    mm64(Ts, 64, Ts, 64, Td, 64, 1.0f, false);
    __syncthreads();
    float* t = Ts; Ts = Td; Td = t;
  }
  if (Ts != bufC) {  // uniform branch: park E in bufC
    for (int e = tid; e < 4096; e += 256) bufC[e] = Ts[e];
    __syncthreads();
  }
  // dA out
  for (int e = tid; e < 4096; e += 256) ws[e] = bufC[e];

  // ---- augmented [deltaA + 1e-5 I | dA - I] --------------------------
  for (int e = tid; e < 4096; e += 256) {
    const int i = e >> 6, j = e & 63;
    const float dtc = fminf(fmaxf(dt[i], -0.1f), 0.1f);
    float m = dtc * (-expf(A_log[e]));
    if (i == j) m += 1e-5f;
    aug[i * 128 + j] = m;
    aug[i * 128 + 64 + j] = bufC[e] - ((i == j) ? 1.0f : 0.0f);
  }
  __syncthreads();

  // ---- Gauss-Jordan with partial pivoting ----------------------------
  for (int col = 0; col < 64; ++col) {
    if (tid == 0) {
      int p = col;
      float best = fabsf(aug[col * 128 + col]);
      for (int r2 = col + 1; r2 < 64; ++r2) {
        float v = fabsf(aug[r2 * 128 + col]);
        if (v > best) { best = v; p = r2; }
      }
      piv_sh = p;
    }
    __syncthreads();
    const int p = piv_sh;
    if (p != col) {
      for (int j = tid; j < 128; j += 256) {
        float t = aug[col * 128 + j];
        aug[col * 128 + j] = aug[p * 128 + j];
        aug[p * 128 + j] = t;
      }
    }
    __syncthreads();
    const float pv = aug[col * 128 + col];
    for (int j = tid; j < 128; j += 256) aug[col * 128 + j] /= pv;
    __syncthreads();
    if (tid < 64) fac[tid] = (tid == col) ? 0.f : aug[tid * 128 + col];
    __syncthreads();
    for (int e = tid; e < 8192; e += 256) {
      const int r2 = e >> 7, j = e & 127;
      aug[e] -= fac[r2] * aug[col * 128 + j];
    }
    __syncthreads();
  }

  // ---- dB = X @ (dtc[:,None]*B), straight to global ------------------
  for (int e = tid; e < 4096; e += 256) {
    const int i = e >> 6;
    const float dtc = fminf(fmaxf(dt[i], -0.1f), 0.1f);
    bufC[e] = dtc * Bm[e];
  }
  __syncthreads();
  mm64(aug + 64, 128, bufC, 64, ws + 4096, 64, 1.0f, false);
}

// Kernel 2: streaming scan. 32768 independent recurrences (b,d,n).
// 128 blocks x 64 threads (2 waves). Thread owns an n-pair (2n, 2n+1) for
// two batches -> 4 independent chains (ILP=4) and b64 NT stores (256 MB of
// h history > 192 MB L2, so bypass).  x staged via async global->LDS b64.
// y computed only at t = SEQ-1.
__global__ __launch_bounds__(64) void ssm_scan(
    const float* __restrict__ x, const float* __restrict__ Cm,
    const float* __restrict__ Dv, const float* __restrict__ ws,
    float* __restrict__ out) {
  const float* dA = ws;
  const float* dB = ws + 4096;
  const int tid = threadIdx.x;      // 64 threads = 2 waves
  const int pl = tid >> 5;          // local d (== wave id)
  const int np = tid & 31;          // n-pair index
  const int dg = blockIdx.x >> 2;   // 0..31 (2 d each)
  const int bg = blockIdx.x & 3;    // 0..3  (batches bg and bg+4)
  const int d = dg * 2 + pl;
  const int d0 = dg * 2;
  const int b1 = bg, b2 = bg + 4;
  const int n0 = np * 2;

  const float da0 = dA[d * 64 + n0], da1 = dA[d * 64 + n0 + 1];
  const float db0 = dB[d * 64 + n0], db1 = dB[d * 64 + n0 + 1];
  const float cv0 = Cm[d * 64 + n0], cv1 = Cm[d * 64 + n0 + 1];

  __shared__ float xs[2][256][2];   // [batch-half][t][local d] : 4 KB
  __shared__ float red[64];

  float h1a = 0.f, h1b = 0.f, h2a = 0.f, h2b = 0.f;
  const size_t o1 = 512 + (size_t)b1 * SEQL * 4096 + (size_t)d * 64 + n0;
  const size_t o2 = 512 + (size_t)b2 * SEQL * 4096 + (size_t)d * 64 + n0;

  for (int t0 = 0; t0 < SEQL; t0 += 256) {
    __syncthreads();                       // previous chunk fully consumed
    for (int rep = 0; rep < 8; ++rep) {    // 512 b64 copies / 64 threads
      const int idx = rep * 64 + tid;
      const int bb = idx >> 8;             // batch half
      const int i = idx & 255;             // t within chunk
      const int bsrc = bb ? b2 : b1;
      stage_b64(x + ((size_t)bsrc * SEQL + t0 + i) * 64 + d0, &xs[bb][i][0]);
    }
    stage_wait();                          // drain my async copies
    __syncthreads();                       // publish LDS to both waves
    if (t0 + 256 < SEQL) {                 // hint next chunk toward L2
      __builtin_prefetch(x + ((size_t)b1 * SEQL + t0 + 256 + tid * 4) * 64 + d0, 0, 0);
      __builtin_prefetch(x + ((size_t)b2 * SEQL + t0 + 256 + tid * 4) * 64 + d0, 0, 0);
    }
    for (int i = 0; i < 256; ++i) {
      const float xv1 = xs[0][i][pl];
      const float xv2 = xs[1][i][pl];
      h1a = fminf(fmaxf(fmaf(xv1, db0, h1a * da0), -1e6f), 1e6f);
      h1b = fminf(fmaxf(fmaf(xv1, db1, h1b * da1), -1e6f), 1e6f);
      h2a = fminf(fmaxf(fmaf(xv2, db0, h2a * da0), -1e6f), 1e6f);
      h2b = fminf(fmaxf(fmaf(xv2, db1, h2b * da1), -1e6f), 1e6f);
      v2f p1; p1.x = h1a; p1.y = h1b;
      v2f p2; p2.x = h2a; p2.y = h2b;
      __builtin_nontemporal_store(p1, (v2f*)(out + o1 + (size_t)(t0 + i) * 4096));
      __builtin_nontemporal_store(p2, (v2f*)(out + o2 + (size_t)(t0 + i) * 4096));
    }
  }

  // y = sum_n h_last * C[d,n] + D[d]*x_last, per (b,d)
  const float xl1 = xs[0][255][pl];
  const float xl2 = xs[1][255][pl];
  red[tid] = h1a * cv0 + h1b * cv1;
  __syncthreads();
  for (int off = 16; off >= 1; off >>= 1) {
    if (np < off) red[tid] += red[tid + off];
    __syncthreads();
  }
  if (np == 0) out[b1 * 64 + d] = red[tid] + Dv[d] * xl1;
  __syncthreads();
  red[tid] = h2a * cv0 + h2b * cv1;
  __syncthreads();
  for (int off = 16; off >= 1; off >>= 1) {
    if (np < off) red[tid] += red[tid + off];
    __syncthreads();
  }
  if (np == 0) out[b2 * 64 + d] = red[tid] + Dv[d] * xl2;
}

extern "C" void kernel_launch(void* const* d_in, const int* in_sizes, int n_in,
                              void* d_out, int out_size, void* d_ws,
                              size_t ws_size, hipStream_t stream) {
  const float* x = (const float*)d_in[0];
  const float* Bm = (const float*)d_in[1];
  const float* Cm = (const float*)d_in[2];
  const float* dt = (const float*)d_in[3];
  const float* Alog = (const float*)d_in[4];
  const float* Dv = (const float*)d_in[5];
  float* out = (float*)d_out;
  float* ws = (float*)d_ws;  // needs 8192 floats (32 KB): dA | dB

  hipLaunchKernelGGL(ssm_setup, dim3(1), dim3(256), 0, stream, Bm, dt, Alog, ws);
  hipLaunchKernelGGL(ssm_scan, dim3(128), dim3(64), 0, stream, x, Cm, Dv, ws, out);
}